// LightCSCF_9689446220002
// MI455X (gfx1250) — compile-verified
//
#include <hip/hip_runtime.h>
#include <hip/hip_bf16.h>
#include <math.h>

typedef __attribute__((ext_vector_type(2))) float v2f;
typedef __attribute__((ext_vector_type(8))) float v8f;

#define BATCH 8192
#define DIM 64
#define INV_T 5.0f      // 1/TEMPERATURE
#define MARGIN 0.8f
#define K_LOG2E 7.21347520444482f          // INV_T * log2(e)
#define MK_LOG2E 5.77078016355585f         // MARGIN * INV_T * log2(e)

// workspace layout (float offsets)
#define WS_F      0                     // 8192*64 normalized user embeddings (e1)
#define WS_G      (BATCH * DIM)         // 8192*64 (e1 + e2)
#define WS_POS    (2 * BATCH * DIM)     // 8192 pos_score
#define WS_ROWSUM (2 * BATCH * DIM + BATCH) // 8192 denominator row sums
#define WS_SCAL   (2 * BATCH * DIM + 2 * BATCH) // [0]=bpr_sum [1]=reg_sum

#define ROWS_PER_BLOCK 128
#define COL_SPLITS 8
#define COLS_PER_SPLIT (BATCH / COL_SPLITS)
#define LDS_PAD 68   // stride in floats: (68 mod 64)=4 spreads ds_read_b64 across banks

// two async global->LDS b128 copies; INST_OFFSET applies to both LDS and mem sides
#define ASYNC_STAGE(ldsoff, src)                                          \
    asm volatile("global_load_async_to_lds_b128 %0, %1, off\n\t"          \
                 "global_load_async_to_lds_b128 %0, %1, off offset:16"    \
                 :: "v"(ldsoff), "v"(src) : "memory")

__global__ void k_init(float* __restrict__ ws) {
    int i = blockIdx.x * blockDim.x + threadIdx.x;
    if (i < BATCH + 16) ws[WS_ROWSUM + i] = 0.0f;  // rowsum[8192] + scalars
}

// one wave32 per batch row: gather, norms, dots, BPR/reg partials, write F/G/pos_score
__global__ void __launch_bounds__(256) k_gather(const float* __restrict__ ut,
                                                const float* __restrict__ it,
                                                const int* __restrict__ user,
                                                const int* __restrict__ posi,
                                                const int* __restrict__ negi,
                                                float* __restrict__ ws) {
    int wave = threadIdx.x >> 5;
    int lane = threadIdx.x & 31;
    int row  = blockIdx.x * 8 + wave;

    int ui = user[row], pi = posi[row], ni = negi[row];
    const float* up  = ut + (size_t)ui * DIM;
    const float* pp  = it + (size_t)pi * DIM;
    const float* npp = it + (size_t)ni * DIM;

    float u0 = up[lane],  u1 = up[lane + 32];
    float p0 = pp[lane],  p1 = pp[lane + 32];
    float n0 = npp[lane], n1 = npp[lane + 32];

    float su  = u0 * u0 + u1 * u1;
    float sp  = p0 * p0 + p1 * p1;
    float sn  = n0 * n0 + n1 * n1;
    float dup = u0 * p0 + u1 * p1;
    float dun = u0 * n0 + u1 * n1;
    #pragma unroll
    for (int s = 16; s > 0; s >>= 1) {   // wave32 butterfly: all lanes get totals
        su  += __shfl_xor(su,  s, 32);
        sp  += __shfl_xor(sp,  s, 32);
        sn  += __shfl_xor(sn,  s, 32);
        dup += __shfl_xor(dup, s, 32);
        dun += __shfl_xor(dun, s, 32);
    }

    float inv_u = 1.0f / fmaxf(sqrtf(su), 1e-12f);
    float inv_p = 1.0f / fmaxf(sqrtf(sp), 1e-12f);

    float* F = ws + WS_F + (size_t)row * DIM;
    float* G = ws + WS_G + (size_t)row * DIM;
    F[lane]      = u0 * inv_u;
    F[lane + 32] = u1 * inv_u;
    G[lane]      = u0 * inv_u + p0 * inv_p;
    G[lane + 32] = u1 * inv_u + p1 * inv_p;

    if (lane == 0) {
        float sim = dup * inv_u * inv_p;
        ws[WS_POS + row] = expf(sim * INV_T) + expf(fmaxf(sim - MARGIN, 0.0f) * INV_T);
        // -log_sigmoid(x) = softplus(-x), numerically stable
        float x  = dup - dun;
        float bl = (x > 0.0f) ? log1pf(expf(-x)) : (-x + log1pf(expf(x)));
        atomicAdd(ws + WS_SCAL + 0, bl);
        atomicAdd(ws + WS_SCAL + 1, su + sp + sn);
    }
}

// S = F @ G^T (8192x8192x64 fp32 WMMA), async double-buffered LDS staging,
// fused exp2 epilogue + row sums
__global__ void __launch_bounds__(256) k_gemm(float* __restrict__ ws) {
    __shared__ float ldsG[2][32 * LDS_PAD];   // double-buffered 32-column stages
    const float* F = ws + WS_F;
    const float* G = ws + WS_G;

    int wave = threadIdx.x >> 5;
    int lane = threadIdx.x & 31;
    int half = lane >> 4;   // 0: lanes 0-15, 1: lanes 16-31
    int l16  = lane & 15;

    int rowbase = blockIdx.x * ROWS_PER_BLOCK + wave * 16;
    int m = rowbase + l16;

    // A fragments: f32 16x4 layout -> lane half selects K pair {k0,k0+1} / {k0+2,k0+3}
    v2f a[16];
    const float* Fm = F + (size_t)m * DIM;
    #pragma unroll
    for (int ks = 0; ks < 16; ++ks) {
        int k = ks * 4 + half * 2;
        a[ks].x = Fm[k];
        a[ks].y = Fm[k + 1];
    }

    float acc[8];
    #pragma unroll
    for (int r = 0; r < 8; ++r) acc[r] = 0.0f;

    // per-thread staging slot: 8 consecutive floats of one column
    int cs = threadIdx.x >> 3;
    int kk = (threadIdx.x & 7) * 8;
    unsigned ldsoff0 = (unsigned)(uintptr_t)&ldsG[0][cs * LDS_PAD + kk];
    unsigned ldsoff1 = ldsoff0 + (unsigned)(32 * LDS_PAD * sizeof(float));
    int cstart = blockIdx.y * COLS_PER_SPLIT;
    const float* gsrc0 = G + (size_t)(cstart + cs) * DIM + kk;

    const int NT = COLS_PER_SPLIT / 32;   // 32 phases of 32 columns

    // prologue: phase 0 -> buffer 0 (ASYNCcnt = 2 for this wave)
    ASYNC_STAGE(ldsoff0, gsrc0);

    for (int ct = 0; ct < NT; ++ct) {
        if (ct + 1 < NT) {
            // prefetch next phase into the other buffer, keep it in flight
            const float* gnext = gsrc0 + (size_t)(ct + 1) * 32 * DIM;
            ASYNC_STAGE((ct & 1) ? ldsoff0 : ldsoff1, gnext);
            // async loads retire in order: cnt<=2 means current buffer landed
            asm volatile("s_wait_asynccnt 0x2" ::: "memory");
        } else {
            asm volatile("s_wait_asynccnt 0x0" ::: "memory");
        }
        __syncthreads();   // all waves' current-buffer data visible in LDS

        const float* buf = &ldsG[ct & 1][0];
        #pragma unroll
        for (int t = 0; t < 2; ++t) {          // two 16-column tiles per phase
            const float* bsrc = buf + (t * 16 + l16) * LDS_PAD;
            v8f c8 = {};
            #pragma unroll
            for (int ks = 0; ks < 16; ++ks) {
                int k = ks * 4 + half * 2;
                v2f b;
                b.x = bsrc[k];
                b.y = bsrc[k + 1];
                c8 = __builtin_amdgcn_wmma_f32_16x16x4_f32(
                    /*neg_a=*/false, a[ks], /*neg_b=*/false, b,
                    /*c_mod=*/(short)0, c8, /*reuse_a=*/false, /*reuse_b=*/false);
            }
            #pragma unroll
            for (int r = 0; r < 8; ++r) {
                // exp(x/T)+exp(relu(x-m)/T) == exp2(x*K) + exp2(max(x*K - m*K, 0))
                float t0 = c8[r] * K_LOG2E;
                acc[r] += __builtin_amdgcn_exp2f(t0)
                        + __builtin_amdgcn_exp2f(fmaxf(t0 - MK_LOG2E, 0.0f));
            }
        }
        __syncthreads();   // everyone done reading this buffer before overwrite
    }

    // C/D layout: VGPR r, lanes 0-15 -> M=r, lanes 16-31 -> M=r+8; N = l16.
    // Sum the 16 column-lanes that share a row, then one atomic per (row, block).
    #pragma unroll
    for (int r = 0; r < 8; ++r) {
        float v = acc[r];
        v += __shfl_xor(v, 1, 32);
        v += __shfl_xor(v, 2, 32);
        v += __shfl_xor(v, 4, 32);
        v += __shfl_xor(v, 8, 32);
        if (l16 == 0) {
            int row = rowbase + r + half * 8;
            atomicAdd(ws + WS_ROWSUM + row, v);
        }
    }
}

__global__ void __launch_bounds__(256) k_final(const float* __restrict__ ws,
                                               float* __restrict__ out) {
    __shared__ float red[256];
    float s = 0.0f;
    for (int i = threadIdx.x; i < BATCH; i += 256) {
        float t = ws[WS_POS + i] / ws[WS_ROWSUM + i] + 1e-5f;
        s += -logf(t);
    }
    red[threadIdx.x] = s;
    __syncthreads();
    for (int st = 128; st > 0; st >>= 1) {
        if (threadIdx.x < st) red[threadIdx.x] += red[threadIdx.x + st];
        __syncthreads();
    }
    if (threadIdx.x == 0) {
        const float nb = 1.0f / (float)BATCH;
        out[0] = ws[WS_SCAL + 0] * nb;                  // bpr
        out[1] = 1e-4f * 0.5f * ws[WS_SCAL + 1] * nb;   // LAMBDA_REG * reg
        out[2] = 0.5f * red[0] * nb;                    // LAMBDA_GAMMA * cscf
    }
}

extern "C" void kernel_launch(void* const* d_in, const int* in_sizes, int n_in,
                              void* d_out, int out_size, void* d_ws, size_t ws_size,
                              hipStream_t stream) {
    const float* ut  = (const float*)d_in[0];
    const float* it  = (const float*)d_in[1];
    const int* user  = (const int*)d_in[2];
    const int* posi  = (const int*)d_in[3];
    const int* negi  = (const int*)d_in[4];
    float* ws  = (float*)d_ws;
    float* out = (float*)d_out;

    hipLaunchKernelGGL(k_init,   dim3((BATCH + 16 + 255) / 256), dim3(256), 0, stream, ws);
    hipLaunchKernelGGL(k_gather, dim3(BATCH / 8), dim3(256), 0, stream,
                       ut, it, user, posi, negi, ws);
    hipLaunchKernelGGL(k_gemm,   dim3(BATCH / ROWS_PER_BLOCK, COL_SPLITS), dim3(256), 0, stream, ws);
    hipLaunchKernelGGL(k_final,  dim3(1), dim3(256), 0, stream, ws, out);
}